// GATv2Net_35167192220487
// MI455X (gfx1250) — compile-verified
//
#include <hip/hip_runtime.h>

#define NN  50000
#define EE  800000
#define EP  (EE + NN)      // 850000 edges incl. self-loops
#define FH  128            // feature width between layers
#define NH  4              // heads
#define HIDW 32            // per-head width

typedef __attribute__((ext_vector_type(2))) float v2f;
typedef __attribute__((ext_vector_type(8))) float v8f;

__device__ __forceinline__ float lrelu(float v) {
    return fmaxf(v, 0.f) + 0.2f * fminf(v, 0.f);
}

// ---------------------------------------------------------------------------
// Fused dual GEMM: xl = X*Wl + bl, xr = X*Wr + br   (X: [M x 128], W: [128 x 128])
// fp32 WMMA 16x16x4. One wave computes the SAME 16x16 tile position for both
// outputs, sharing the A fragment -> half the A traffic and two independent
// WMMA accumulator chains (hides the WMMA->WMMA RAW latency).
// 8 waves per block cover 128 rows x 16 cols.
// ---------------------------------------------------------------------------
__global__ void gemm_wmma_dual(const float* __restrict__ X,
                               const float* __restrict__ Wl,
                               const float* __restrict__ bl,
                               const float* __restrict__ Wr,
                               const float* __restrict__ br,
                               float* __restrict__ OutL,
                               float* __restrict__ OutR)
{
    const int lane  = threadIdx.x & 31;
    const int wave  = threadIdx.x >> 5;
    const int r0    = blockIdx.x * 128 + wave * 16;   // C tile row base
    const int n0    = blockIdx.y * 16;                // C tile col base
    const int mrow  = lane & 15;                      // M (or N) within tile
    const int khalf = lane >> 4;                      // 0 -> K{0,1}, 1 -> K{2,3}

    int arow = r0 + mrow;
    if (arow >= NN) arow = NN - 1;                    // clamp (EXEC must stay all-1 for WMMA)
    const float* xrow = X + (size_t)arow * FH;
    const int col = n0 + mrow;

    v8f cL = {};
    v8f cR = {};
#pragma unroll
    for (int k = 0; k < FH; k += 4) {
        // A 16x4 fragment: lanes 0-15 hold K=k,k+1 ; lanes 16-31 hold K=k+2,k+3
        const v2f a = *(const v2f*)(xrow + k + khalf * 2);
        // B 4x16 fragments: row K striped across lanes within a VGPR
        const size_t rk0 = (size_t)(k + khalf * 2) * FH + col;
        v2f bL, bR;
        bL.x = Wl[rk0];       bL.y = Wl[rk0 + FH];
        bR.x = Wr[rk0];       bR.y = Wr[rk0 + FH];
        cL = __builtin_amdgcn_wmma_f32_16x16x4_f32(false, a, false, bL,
                                                   (short)0, cL, false, false);
        cR = __builtin_amdgcn_wmma_f32_16x16x4_f32(false, a, false, bR,
                                                   (short)0, cR, false, false);
    }

    const float bbL = bl[col];
    const float bbR = br[col];
#pragma unroll
    for (int v = 0; v < 8; ++v) {
        const int row = r0 + v + khalf * 8;           // C layout: VGPR v -> M=v / M=8+v
        if (row < NN) {
            OutL[(size_t)row * FH + col] = cL[v] + bbL;
            OutR[(size_t)row * FH + col] = cR[v] + bbR;
        }
    }
}

// ---------------------------------------------------------------------------
// Zero the accumulator h, init segment-max to -inf and denom to 0.
// ---------------------------------------------------------------------------
__global__ void init_buffers(float* __restrict__ hbuf,
                             float* __restrict__ m,
                             float* __restrict__ denom)
{
    const int t = blockIdx.x * blockDim.x + threadIdx.x;
    if (t < NN * FH) hbuf[t] = 0.f;
    if (t < NN * NH) { m[t] = -3.0e38f; denom[t] = 0.f; }
}

// ---------------------------------------------------------------------------
// Pass 1: per-edge attention logits + segment max (atomic fp32 max).
// One wave per edge; lane handles 4 features; 8 lanes per head.
// ---------------------------------------------------------------------------
__global__ void edge_logits(const float* __restrict__ xl,
                            const float* __restrict__ xr,
                            const int* __restrict__ src,
                            const int* __restrict__ dst,
                            const float* __restrict__ att,   // [NH*HIDW] flat
                            float* __restrict__ lg,          // [EP*NH]
                            float* __restrict__ m)           // [NN*NH]
{
    const int gid  = blockIdx.x * blockDim.x + threadIdx.x;
    const int e    = gid >> 5;
    const int lane = gid & 31;
    if (e >= EP) return;

    int s, d;
    if (e < EE) { s = src[e]; d = dst[e]; } else { s = d = e - EE; }

    const int f = lane * 4;
    const float4 a = *(const float4*)(xl + (size_t)s * FH + f);
    const float4 b = *(const float4*)(xr + (size_t)d * FH + f);
    const float4 w = *(const float4*)(att + f);

    float p = lrelu(a.x + b.x) * w.x + lrelu(a.y + b.y) * w.y +
              lrelu(a.z + b.z) * w.z + lrelu(a.w + b.w) * w.w;
    // reduce across the 8 lanes of one head
    p += __shfl_xor(p, 1, 8);
    p += __shfl_xor(p, 2, 8);
    p += __shfl_xor(p, 4, 8);

    if ((lane & 7) == 0) {
        const int h = lane >> 3;
        lg[(size_t)e * NH + h] = p;
        atomicMax(&m[(size_t)d * NH + h], p);
    }
}

// ---------------------------------------------------------------------------
// Pass 2: ex = exp(logit - m[dst]); denom[dst] += ex.
// One thread per edge handles all 4 heads (shared dst load, float4 lg access).
// ---------------------------------------------------------------------------
__global__ void edge_exp(const int* __restrict__ dst,
                         const float* __restrict__ m,
                         float* __restrict__ lg,
                         float* __restrict__ denom)
{
    const int e = blockIdx.x * blockDim.x + threadIdx.x;
    if (e >= EP) return;
    const int d = (e < EE) ? dst[e] : (e - EE);

    float4 l  = *(const float4*)(lg + (size_t)e * NH);
    const float4 mm = *(const float4*)(m + (size_t)d * NH);
    l.x = expf(l.x - mm.x);
    l.y = expf(l.y - mm.y);
    l.z = expf(l.z - mm.z);
    l.w = expf(l.w - mm.w);
    *(float4*)(lg + (size_t)e * NH) = l;

    float* dn = denom + (size_t)d * NH;
    atomicAdd(dn + 0, l.x);
    atomicAdd(dn + 1, l.y);
    atomicAdd(dn + 2, l.z);
    atomicAdd(dn + 3, l.w);
}

// ---------------------------------------------------------------------------
// Pass 3: h[dst] += alpha * xl[src].  One wave per edge, float4 per lane,
// fp32 global atomic adds.
// ---------------------------------------------------------------------------
__global__ void edge_scatter(const float* __restrict__ xl,
                             const int* __restrict__ src,
                             const int* __restrict__ dst,
                             const float* __restrict__ lg,
                             const float* __restrict__ denom,
                             float* __restrict__ acc)
{
    const int gid  = blockIdx.x * blockDim.x + threadIdx.x;
    const int e    = gid >> 5;
    const int lane = gid & 31;
    if (e >= EP) return;

    int s, d;
    if (e < EE) { s = src[e]; d = dst[e]; } else { s = d = e - EE; }

    const int h = lane >> 3;
    const float alpha = lg[(size_t)e * NH + h] /
                        fmaxf(denom[(size_t)d * NH + h], 1e-16f);

    const int f = lane * 4;
    const float4 a = *(const float4*)(xl + (size_t)s * FH + f);
    float* o = acc + (size_t)d * FH + f;
    atomicAdd(o + 0, a.x * alpha);
    atomicAdd(o + 1, a.y * alpha);
    atomicAdd(o + 2, a.z * alpha);
    atomicAdd(o + 3, a.w * alpha);
}

// ---------------------------------------------------------------------------
// h = elu(h + bo), in place.
// ---------------------------------------------------------------------------
__global__ void finalize_elu(float* __restrict__ hbuf,
                             const float* __restrict__ bo)
{
    const int t = blockIdx.x * blockDim.x + threadIdx.x;
    if (t >= NN * FH) return;
    const float v = hbuf[t] + bo[t & (FH - 1)];
    hbuf[t] = (v > 0.f) ? v : (expf(v) - 1.f);
}

// ---------------------------------------------------------------------------
// out[n] = h[n,:] . Wf + bf.  One wave per node.
// ---------------------------------------------------------------------------
__global__ void final_linear(const float* __restrict__ h,
                             const float* __restrict__ Wf,
                             const float* __restrict__ bf,
                             float* __restrict__ out)
{
    const int gid  = blockIdx.x * blockDim.x + threadIdx.x;
    const int n    = gid >> 5;
    const int lane = gid & 31;
    if (n >= NN) return;

    const int f = lane * 4;
    const float4 a = *(const float4*)(h + (size_t)n * FH + f);
    const float4 w = *(const float4*)(Wf + f);
    float p = a.x * w.x + a.y * w.y + a.z * w.z + a.w * w.w;
    p += __shfl_xor(p, 16, 32);
    p += __shfl_xor(p,  8, 32);
    p += __shfl_xor(p,  4, 32);
    p += __shfl_xor(p,  2, 32);
    p += __shfl_xor(p,  1, 32);
    if (lane == 0) out[n] = p + bf[0];
}

// ---------------------------------------------------------------------------
extern "C" void kernel_launch(void* const* d_in, const int* in_sizes, int n_in,
                              void* d_out, int out_size, void* d_ws, size_t ws_size,
                              hipStream_t stream)
{
    const float* x    = (const float*)d_in[0];
    const int*   ei   = (const int*)d_in[1];
    const int*   src  = ei;
    const int*   dst  = ei + EE;
    const float* Wl0  = (const float*)d_in[2];
    const float* bl0  = (const float*)d_in[3];
    const float* Wr0  = (const float*)d_in[4];
    const float* br0  = (const float*)d_in[5];
    const float* att0 = (const float*)d_in[6];
    const float* bo0  = (const float*)d_in[7];
    const float* Wl   = (const float*)d_in[8];
    const float* bl   = (const float*)d_in[9];
    const float* Wr   = (const float*)d_in[10];
    const float* br   = (const float*)d_in[11];
    const float* att  = (const float*)d_in[12];
    const float* bo   = (const float*)d_in[13];
    const float* Wf   = (const float*)d_in[14];
    const float* bf   = (const float*)d_in[15];

    float* ws   = (float*)d_ws;
    float* hbuf = ws;                              // NN*FH
    float* xl   = hbuf + (size_t)NN * FH;          // NN*FH
    float* xr   = xl   + (size_t)NN * FH;          // NN*FH
    float* lg   = xr   + (size_t)NN * FH;          // EP*NH
    float* mbuf = lg   + (size_t)EP * NH;          // NN*NH
    float* dbuf = mbuf + (size_t)NN * NH;          // NN*NH

    const dim3 gemmGrid((NN + 127) / 128, FH / 16);
    const dim3 edgeGrid((EP * 32 + 255) / 256);
    const dim3 eGrid((EP + 255) / 256);
    const dim3 nodeGrid((NN * FH + 255) / 256);
    const dim3 linGrid((NN * 32 + 255) / 256);

    for (int layer = 0; layer < 3; ++layer) {
        const float* in = (layer == 0) ? x : hbuf;
        const float *WL, *BL, *WR, *BR, *AT, *BO;
        if (layer == 0) { WL = Wl0; BL = bl0; WR = Wr0; BR = br0; AT = att0; BO = bo0; }
        else {
            const int i = layer - 1;
            WL = Wl + (size_t)i * FH * FH;  BL = bl + (size_t)i * FH;
            WR = Wr + (size_t)i * FH * FH;  BR = br + (size_t)i * FH;
            AT = att + (size_t)i * NH * HIDW;
            BO = bo + (size_t)i * FH;
        }
        gemm_wmma_dual<<<gemmGrid, 256, 0, stream>>>(in, WL, BL, WR, BR, xl, xr);
        init_buffers <<<nodeGrid, 256, 0, stream>>>(hbuf, mbuf, dbuf);
        edge_logits  <<<edgeGrid, 256, 0, stream>>>(xl, xr, src, dst, AT, lg, mbuf);
        edge_exp     <<<eGrid,    256, 0, stream>>>(dst, mbuf, lg, dbuf);
        edge_scatter <<<edgeGrid, 256, 0, stream>>>(xl, src, dst, lg, dbuf, hbuf);
        finalize_elu <<<nodeGrid, 256, 0, stream>>>(hbuf, BO);
    }
    final_linear<<<linGrid, 256, 0, stream>>>(hbuf, Wf, bf, (float*)d_out);
}